// SAGEConvNet_34110630265037
// MI455X (gfx1250) — compile-verified
//
#include <hip/hip_runtime.h>
#include <hip/hip_bf16.h>
#include <math.h>

#define N_NODES 100000
#define N_EDGES 1600000
#define HIDDEN  128
#define N_GRAPHS 64
#define N_CLASSES 10

typedef __attribute__((ext_vector_type(2))) float v2f;
typedef __attribute__((ext_vector_type(8))) float v8f;

// ---------------------------------------------------------------------------
// Degree (in-degree per node, as float) and per-graph node counts
// ---------------------------------------------------------------------------
__global__ void deg_kernel(const int* __restrict__ ei, float* __restrict__ deg, int E) {
    int e = blockIdx.x * blockDim.x + threadIdx.x;
    if (e < E) atomicAdd(deg + ei[E + e], 1.0f);   // dst = ei[1][e]
}

__global__ void gcnt_kernel(const int* __restrict__ batch, float* __restrict__ gcnt, int N) {
    int v = blockIdx.x * blockDim.x + threadIdx.x;
    if (v < N) atomicAdd(gcnt + batch[v], 1.0f);
}

// ---------------------------------------------------------------------------
// WMMA fp32 GEMM:  C[M,128] = A[M,128] @ W[128,128]^T  (+C) (+bias)
// One wave computes one 16x16 tile, K swept in steps of 4 via
// V_WMMA_F32_16X16X4_F32. Block = 256 threads = 8 waves = one 16-row stripe
// across all 8 N-tiles. M must be a multiple of 16 (100000 = 6250*16).
//
// A fragment (16x4, 32-bit): lane m = l&15; lanes 0-15 hold K=k+0..1,
// lanes 16-31 hold K=k+2..3 (2 VGPRs / lane -> float2).
// B fragment (4x16): lane n = l&15; same K split across lane halves.
// C/D (16x16 f32): VGPR v, lanes 0-15 -> M=v, lanes 16-31 -> M=v+8, N=l&15.
// ---------------------------------------------------------------------------
template<bool ACC, bool BIAS>
__global__ void gemm_nt_wmma(const float* __restrict__ A,
                             const float* __restrict__ W,
                             const float* __restrict__ bias,
                             float* __restrict__ C, int M) {
    const int wave = threadIdx.x >> 5;           // 0..7 -> N tile
    const int lane = threadIdx.x & 31;
    const int row0 = blockIdx.x * 16;            // M tile base
    const int n0   = wave * 16;
    const int mi   = lane & 15;
    const int koff = (lane >> 4) << 1;           // 0 or 2

    const float* __restrict__ Ap = A + (size_t)(row0 + mi) * HIDDEN + koff;
    const float* __restrict__ Wp = W + (size_t)(n0 + mi) * HIDDEN + koff; // B[k][n]=W[n][k]

    v8f acc = {};
#pragma unroll 4
    for (int k = 0; k < HIDDEN; k += 4) {
        v2f a = *(const v2f*)(Ap + k);
        v2f b = *(const v2f*)(Wp + k);
        acc = __builtin_amdgcn_wmma_f32_16x16x4_f32(
            /*neg_a=*/false, a, /*neg_b=*/false, b,
            /*c_mod=*/(short)0, acc, /*reuse_a=*/false, /*reuse_b=*/false);
    }

    const int n  = lane & 15;
    const int mb = (lane >> 4) << 3;             // 0 or 8
    float* __restrict__ Cp = C + (size_t)(row0 + mb) * HIDDEN + n0 + n;
    const float bv = BIAS ? bias[n0 + n] : 0.0f;
#pragma unroll
    for (int v = 0; v < 8; ++v) {
        float r = acc[v] + bv;
        if (ACC) r += Cp[(size_t)v * HIDDEN];
        Cp[(size_t)v * HIDDEN] = r;
    }
}

// ---------------------------------------------------------------------------
// Edge scatter kernels: one wave per edge, lane covers 4 floats (float4 gather
// + 4 global_atomic_add_f32). 51MB feature matrix fits in 192MB L2, so these
// are L2-resident.
// ---------------------------------------------------------------------------
__global__ void gcn_edge_kernel(const int* __restrict__ ei, const float* __restrict__ deg,
                                const float* __restrict__ xw, float* __restrict__ out, int E) {
    int e = blockIdx.x * (blockDim.x >> 5) + (threadIdx.x >> 5);
    if (e >= E) return;
    const int lane = threadIdx.x & 31;
    const int s = ei[e], d = ei[E + e];
    const float nrm = rsqrtf((deg[s] + 1.0f) * (deg[d] + 1.0f));
    float4 v = ((const float4*)(xw + (size_t)s * HIDDEN))[lane];
    float* dp = out + (size_t)d * HIDDEN + lane * 4;
    atomicAdd(dp + 0, v.x * nrm);
    atomicAdd(dp + 1, v.y * nrm);
    atomicAdd(dp + 2, v.z * nrm);
    atomicAdd(dp + 3, v.w * nrm);
}

__global__ void sage_edge_kernel(const int* __restrict__ ei, const float* __restrict__ h,
                                 float* __restrict__ agg, int E) {
    int e = blockIdx.x * (blockDim.x >> 5) + (threadIdx.x >> 5);
    if (e >= E) return;
    const int lane = threadIdx.x & 31;
    const int s = ei[e], d = ei[E + e];
    float4 v = ((const float4*)(h + (size_t)s * HIDDEN))[lane];
    float* dp = agg + (size_t)d * HIDDEN + lane * 4;
    atomicAdd(dp + 0, v.x);
    atomicAdd(dp + 1, v.y);
    atomicAdd(dp + 2, v.z);
    atomicAdd(dp + 3, v.w);
}

// h = agg + xw/(deg+1) + b      (GCN self-loop term + bias)
__global__ void gcn_self_kernel(const float* __restrict__ agg, const float* __restrict__ xw,
                                const float* __restrict__ deg, const float* __restrict__ b,
                                float* __restrict__ h, int N) {
    long i = (long)blockIdx.x * blockDim.x + threadIdx.x;
    if (i >= (long)N * HIDDEN) return;
    int v = (int)(i >> 7), f = (int)(i & 127);
    h[i] = agg[i] + xw[i] / (deg[v] + 1.0f) + b[f];
}

// agg /= max(deg, 1)
__global__ void sage_scale_kernel(float* __restrict__ agg, const float* __restrict__ deg, int N) {
    long i = (long)blockIdx.x * blockDim.x + threadIdx.x;
    if (i >= (long)N * HIDDEN) return;
    int v = (int)(i >> 7);
    agg[i] /= fmaxf(deg[v], 1.0f);
}

// ---------------------------------------------------------------------------
// Global mean pool
// ---------------------------------------------------------------------------
__global__ void pool_add_kernel(const float* __restrict__ h, const int* __restrict__ batch,
                                float* __restrict__ g, int N) {
    int v = blockIdx.x * (blockDim.x >> 5) + (threadIdx.x >> 5);
    if (v >= N) return;
    const int lane = threadIdx.x & 31;
    const int gi = batch[v];
    float4 x = ((const float4*)(h + (size_t)v * HIDDEN))[lane];
    float* gp = g + (size_t)gi * HIDDEN + lane * 4;
    atomicAdd(gp + 0, x.x);
    atomicAdd(gp + 1, x.y);
    atomicAdd(gp + 2, x.z);
    atomicAdd(gp + 3, x.w);
}

__global__ void pool_div_kernel(float* __restrict__ g, const float* __restrict__ gcnt) {
    int i = blockIdx.x * blockDim.x + threadIdx.x;
    if (i >= N_GRAPHS * HIDDEN) return;
    g[i] /= fmaxf(gcnt[i >> 7], 1.0f);
}

// ---------------------------------------------------------------------------
// Classifier head: out = [tanh(bn(] in @ W^T + bias [))]. Tiny (<=64x200).
// ---------------------------------------------------------------------------
__global__ void fc_kernel(const float* __restrict__ in, const float* __restrict__ W,
                          const float* __restrict__ bias, const float* __restrict__ bng,
                          const float* __restrict__ bnb, float* __restrict__ out,
                          int B, int In, int Out, int act) {
    int i = blockIdx.x * blockDim.x + threadIdx.x;
    if (i >= B * Out) return;
    int b = i / Out, o = i % Out;
    const float* ip = in + (size_t)b * In;
    const float* wp = W + (size_t)o * In;
    float s = bias[o];
    for (int k = 0; k < In; ++k) s += ip[k] * wp[k];
    if (act) {
        const float bnscale = 0.999995000037499f;   // 1/sqrt(1 + 1e-5)
        s = tanhf(s * (bng[o] * bnscale) + bnb[o]);
    }
    out[i] = s;
}

__global__ void softmax_kernel(const float* __restrict__ logits, float* __restrict__ out) {
    int b = blockIdx.x * blockDim.x + threadIdx.x;
    if (b >= N_GRAPHS) return;
    const float* lp = logits + b * N_CLASSES;
    float mx = lp[0];
    for (int c = 1; c < N_CLASSES; ++c) mx = fmaxf(mx, lp[c]);
    float sum = 0.0f, e[N_CLASSES];
    for (int c = 0; c < N_CLASSES; ++c) { e[c] = __expf(lp[c] - mx); sum += e[c]; }
    float inv = 1.0f / sum;
    for (int c = 0; c < N_CLASSES; ++c) out[b * N_CLASSES + c] = e[c] * inv;
}

// ---------------------------------------------------------------------------
extern "C" void kernel_launch(void* const* d_in, const int* in_sizes, int n_in,
                              void* d_out, int out_size, void* d_ws, size_t ws_size,
                              hipStream_t stream) {
    const float* x       = (const float*)d_in[0];
    const int*   ei      = (const int*)  d_in[1];   // [2, E]
    const int*   batch   = (const int*)  d_in[2];
    const float* gcn_w   = (const float*)d_in[3];
    const float* gcn_b   = (const float*)d_in[4];
    const float* s1_wl   = (const float*)d_in[5];
    const float* s1_bl   = (const float*)d_in[6];
    const float* s1_wr   = (const float*)d_in[7];
    const float* s2_wl   = (const float*)d_in[8];
    const float* s2_bl   = (const float*)d_in[9];
    const float* s2_wr   = (const float*)d_in[10];
    const float* w0 = (const float*)d_in[11]; const float* b0 = (const float*)d_in[12];
    const float* w1 = (const float*)d_in[13]; const float* b1 = (const float*)d_in[14];
    const float* w2 = (const float*)d_in[15]; const float* b2 = (const float*)d_in[16];
    const float* w3 = (const float*)d_in[17]; const float* b3 = (const float*)d_in[18];
    const float* bn0g = (const float*)d_in[19]; const float* bn0b = (const float*)d_in[20];
    const float* bn1g = (const float*)d_in[21]; const float* bn1b = (const float*)d_in[22];
    const float* bn2g = (const float*)d_in[23]; const float* bn2b = (const float*)d_in[24];

    const size_t NH = (size_t)N_NODES * HIDDEN;
    float* bufA = (float*)d_ws;          // h buffers
    float* bufB = bufA + NH;             // aggregation accumulator
    float* bufC = bufB + NH;             // linear output
    float* deg  = bufC + NH;             // [N] in-degree (float)
    float* gsum = deg + N_NODES;         // [64,128]
    float* gcnt = gsum + N_GRAPHS * HIDDEN;
    float* t0 = gcnt + N_GRAPHS;         // [64,200]
    float* t1 = t0 + N_GRAPHS * 200;     // [64,100]
    float* t2 = t1 + N_GRAPHS * 100;     // [64,50]
    float* lg = t2 + N_GRAPHS * 50;      // [64,10]

    const int E = N_EDGES, N = N_NODES;
    const int gemmBlocks = N / 16;                     // 6250 (exact)
    const int edgeBlocks = (E + 7) / 8;                // 8 edges / 256-thr block
    const int elemBlocks = (int)((NH + 255) / 256);
    const int nodeBlocks = (N + 7) / 8;

    // degrees + graph counts
    hipMemsetAsync(deg,  0, (size_t)N_NODES * sizeof(float), stream);
    hipMemsetAsync(gcnt, 0, N_GRAPHS * sizeof(float), stream);
    deg_kernel<<<(E + 255) / 256, 256, 0, stream>>>(ei, deg, E);
    gcnt_kernel<<<(N + 255) / 256, 256, 0, stream>>>(batch, gcnt, N);

    // ---- GCNConv: xw = x @ gcn_w^T; agg = sum norm*xw[src]; h0 = agg + xw/(deg+1) + b
    gemm_nt_wmma<false, false><<<gemmBlocks, 256, 0, stream>>>(x, gcn_w, nullptr, bufC, N);
    hipMemsetAsync(bufB, 0, NH * sizeof(float), stream);
    gcn_edge_kernel<<<edgeBlocks, 256, 0, stream>>>(ei, deg, bufC, bufB, E);
    gcn_self_kernel<<<elemBlocks, 256, 0, stream>>>(bufB, bufC, deg, gcn_b, bufA, N);

    // ---- SAGE1: h1 = mean_agg(h0) @ wl^T + bl + h0 @ wr^T
    hipMemsetAsync(bufB, 0, NH * sizeof(float), stream);
    sage_edge_kernel<<<edgeBlocks, 256, 0, stream>>>(ei, bufA, bufB, E);
    sage_scale_kernel<<<elemBlocks, 256, 0, stream>>>(bufB, deg, N);
    gemm_nt_wmma<false, true><<<gemmBlocks, 256, 0, stream>>>(bufB, s1_wl, s1_bl, bufC, N);
    gemm_nt_wmma<true,  false><<<gemmBlocks, 256, 0, stream>>>(bufA, s1_wr, nullptr, bufC, N);

    // ---- SAGE2: h2 = mean_agg(h1) @ wl^T + bl + h1 @ wr^T
    hipMemsetAsync(bufB, 0, NH * sizeof(float), stream);
    sage_edge_kernel<<<edgeBlocks, 256, 0, stream>>>(ei, bufC, bufB, E);
    sage_scale_kernel<<<elemBlocks, 256, 0, stream>>>(bufB, deg, N);
    gemm_nt_wmma<false, true><<<gemmBlocks, 256, 0, stream>>>(bufB, s2_wl, s2_bl, bufA, N);
    gemm_nt_wmma<true,  false><<<gemmBlocks, 256, 0, stream>>>(bufC, s2_wr, nullptr, bufA, N);

    // ---- global mean pool
    hipMemsetAsync(gsum, 0, N_GRAPHS * HIDDEN * sizeof(float), stream);
    pool_add_kernel<<<nodeBlocks, 256, 0, stream>>>(bufA, batch, gsum, N);
    pool_div_kernel<<<(N_GRAPHS * HIDDEN + 255) / 256, 256, 0, stream>>>(gsum, gcnt);

    // ---- classifier head
    fc_kernel<<<(N_GRAPHS * 200 + 255) / 256, 256, 0, stream>>>(gsum, w0, b0, bn0g, bn0b, t0, N_GRAPHS, 128, 200, 1);
    fc_kernel<<<(N_GRAPHS * 100 + 255) / 256, 256, 0, stream>>>(t0, w1, b1, bn1g, bn1b, t1, N_GRAPHS, 200, 100, 1);
    fc_kernel<<<(N_GRAPHS *  50 + 255) / 256, 256, 0, stream>>>(t1, w2, b2, bn2g, bn2b, t2, N_GRAPHS, 100, 50, 1);
    fc_kernel<<<(N_GRAPHS *  10 + 255) / 256, 256, 0, stream>>>(t2, w3, b3, nullptr, nullptr, lg, N_GRAPHS, 50, 10, 0);
    softmax_kernel<<<1, 64, 0, stream>>>(lg, (float*)d_out);
}